// autoencoder_20426864460458
// MI455X (gfx1250) — compile-verified
//
#include <hip/hip_runtime.h>
#include <math.h>
#include <stdint.h>

typedef __attribute__((ext_vector_type(2))) float v2f;
typedef __attribute__((ext_vector_type(8))) float v8f;

#define BATCH 2048
#define A_DIM 6144
#define D_DIM 4096

// ---------------------------------------------------------------------------
// GEMV layer: out[row] = act( W[row,:] . x + bias[row] )
// W is (fan_out, fan_in) row-major.  Uses V_WMMA_F32_16X16X4_F32:
//   A = 16x4 tile of W (lane m=lane&15 holds row m; VGPR pair holds K=2h,2h+1)
//   B = 4x16 with x[k..k+3] in column 0 only  ->  D[:,0] += W_tile . x_chunk
// Block = 256 threads = 8 waves: 2 row-groups (16 rows each) x 4 K-slices.
// x is staged into LDS with GLOBAL_LOAD_ASYNC_TO_LDS_B128 (ASYNCcnt path).
// ACT: 0 = none, 1 = swish, 2 = tanh
// ---------------------------------------------------------------------------
template <int ACT>
__global__ __launch_bounds__(256) void gemv_wmma_kernel(
    const float* __restrict__ W, const float* __restrict__ bias,
    const float* __restrict__ xin, float* __restrict__ out, int fan_in)
{
    __shared__ float xs[A_DIM];        // input vector (<= 24 KB)
    __shared__ float part[2][4][16];   // [row-group][k-slice][row]

    const int tid = threadIdx.x;

    // ---- async stage x -> LDS (16B per lane per trip; trip count is uniform:
    //      fan_in/4/256 = 6 (6144) or 4 (4096), so EXEC is all-ones throughout)
    const int n4 = fan_in >> 2;
    for (int i = tid; i < n4; i += 256) {
        uint32_t lds_dst = (uint32_t)(uintptr_t)(xs + 4 * i);   // low 32b = LDS addr
        uint64_t gsrc    = (uint64_t)(uintptr_t)(xin + 4 * i);
        asm volatile("global_load_async_to_lds_b128 %0, %1, off"
                     :: "v"(lds_dst), "v"(gsrc) : "memory");
    }
    asm volatile("s_wait_asynccnt 0x0" ::: "memory");
    __syncthreads();

    const int wave = tid >> 5;
    const int lane = tid & 31;
    const int m    = lane & 15;   // row within 16-row tile
    const int h    = lane >> 4;   // lane-half: K offset {0,2}
    const int rg   = wave & 1;    // row group within block
    const int ks   = wave >> 1;   // K slice 0..3

    const int klen = fan_in >> 2; // per-slice K length (1536 or 1024)
    const int k0   = ks * klen;

    const int row  = blockIdx.x * 32 + rg * 16 + m;
    const float* wp = W + (size_t)row * fan_in + k0 + 2 * h;
    const float* xq = xs + k0 + 2 * h;
    const bool bact = (m == 0);   // only column N=0 of B is live

    v8f c = {0.f, 0.f, 0.f, 0.f, 0.f, 0.f, 0.f, 0.f};

    for (int k = 0; k < klen; k += 16) {
        v2f a0 = *(const v2f*)(wp + k);
        v2f a1 = *(const v2f*)(wp + k + 4);
        v2f a2 = *(const v2f*)(wp + k + 8);
        v2f a3 = *(const v2f*)(wp + k + 12);
        v2f b0, b1, b2, b3;
        b0[0] = bact ? xq[k]      : 0.f;  b0[1] = bact ? xq[k + 1]  : 0.f;
        b1[0] = bact ? xq[k + 4]  : 0.f;  b1[1] = bact ? xq[k + 5]  : 0.f;
        b2[0] = bact ? xq[k + 8]  : 0.f;  b2[1] = bact ? xq[k + 9]  : 0.f;
        b3[0] = bact ? xq[k + 12] : 0.f;  b3[1] = bact ? xq[k + 13] : 0.f;
        c = __builtin_amdgcn_wmma_f32_16x16x4_f32(false, a0, false, b0,
                                                  (short)0, c, false, false);
        c = __builtin_amdgcn_wmma_f32_16x16x4_f32(false, a1, false, b1,
                                                  (short)0, c, false, false);
        c = __builtin_amdgcn_wmma_f32_16x16x4_f32(false, a2, false, b2,
                                                  (short)0, c, false, false);
        c = __builtin_amdgcn_wmma_f32_16x16x4_f32(false, a3, false, b3,
                                                  (short)0, c, false, false);
        __builtin_prefetch(wp + k + 512, 0, 0);  // global_prefetch_b8, ~2KB ahead
    }

    // D layout: VGPR j -> (lane 0: row j), (lane 16: row 8+j), column N = 0
    if (m == 0) {
#pragma unroll
        for (int j = 0; j < 8; ++j)
            part[rg][ks][h * 8 + j] = c[j];
    }
    __syncthreads();

    if (tid < 32) {
        const int rg2 = tid >> 4, m2 = tid & 15;
        float v = part[rg2][0][m2] + part[rg2][1][m2] +
                  part[rg2][2][m2] + part[rg2][3][m2];
        const int orow = blockIdx.x * 32 + tid;
        v += bias[orow];
        if (ACT == 1)      v = v * (1.f / (1.f + __expf(-v)));
        else if (ACT == 2) v = tanhf(v);
        out[orow] = v;
    }
}

// ---------------------------------------------------------------------------
// KL-distance tail: centroids -> per-row L2 distances -> log_softmax -> KL sum
// Single 256-thread block; all data (2 x 2048 floats) lives in LDS.
// ---------------------------------------------------------------------------
__device__ inline float block_sum(float v, float* red) {
    int tid = threadIdx.x;
    red[tid] = v; __syncthreads();
    for (int s = 128; s > 0; s >>= 1) {
        if (tid < s) red[tid] += red[tid + s];
        __syncthreads();
    }
    float r = red[0]; __syncthreads();
    return r;
}

__device__ inline float block_max(float v, float* red) {
    int tid = threadIdx.x;
    red[tid] = v; __syncthreads();
    for (int s = 128; s > 0; s >>= 1) {
        if (tid < s) red[tid] = fmaxf(red[tid], red[tid + s]);
        __syncthreads();
    }
    float r = red[0]; __syncthreads();
    return r;
}

__global__ __launch_bounds__(256) void distance_kernel(
    const float* __restrict__ x, const float* __restrict__ lat,
    float* __restrict__ out)
{
    __shared__ float din[BATCH];
    __shared__ float dlat[BATCH];
    __shared__ float red[256];
    const int tid = threadIdx.x;

    // centroids
    float s0 = 0.f, s1 = 0.f, s2 = 0.f, t0 = 0.f, t1 = 0.f;
    for (int b = tid; b < BATCH; b += 256) {
        s0 += x[3 * b];     s1 += x[3 * b + 1]; s2 += x[3 * b + 2];
        t0 += lat[2 * b];   t1 += lat[2 * b + 1];
    }
    const float inv = 1.f / (float)BATCH;
    float c0 = block_sum(s0, red) * inv;
    float c1 = block_sum(s1, red) * inv;
    float c2 = block_sum(s2, red) * inv;
    float l0 = block_sum(t0, red) * inv;
    float l1 = block_sum(t1, red) * inv;

    // per-row distances to centroid
    for (int b = tid; b < BATCH; b += 256) {
        float d0 = x[3 * b] - c0, d1 = x[3 * b + 1] - c1, d2 = x[3 * b + 2] - c2;
        din[b]  = sqrtf(d0 * d0 + d1 * d1 + d2 * d2);
        float e0 = lat[2 * b] - l0, e1 = lat[2 * b + 1] - l1;
        dlat[b] = sqrtf(e0 * e0 + e1 * e1);
    }
    __syncthreads();

    // stable log_softmax over both 2048-vectors
    float mp = -INFINITY, mq = -INFINITY;
    for (int b = tid; b < BATCH; b += 256) {
        mp = fmaxf(mp, din[b]);
        mq = fmaxf(mq, dlat[b]);
    }
    mp = block_max(mp, red);
    mq = block_max(mq, red);

    float sp = 0.f, sq = 0.f;
    for (int b = tid; b < BATCH; b += 256) {
        sp += __expf(din[b]  - mp);
        sq += __expf(dlat[b] - mq);
    }
    sp = block_sum(sp, red);
    sq = block_sum(sq, red);
    const float lsp = logf(sp), lsq = logf(sq);

    // KL(q || p) = sum q * (log q - log p)
    float kl = 0.f;
    for (int b = tid; b < BATCH; b += 256) {
        float lp = din[b]  - mp - lsp;
        float lq = dlat[b] - mq - lsq;
        kl += __expf(lq) * (lq - lp);
    }
    kl = block_sum(kl, red);
    if (tid == 0) out[0] = kl;
}

// ---------------------------------------------------------------------------
// d_in layout: [0]=x, [1..8]=enc (W,b)x4, [9..16]=dec (W,b)x4
// d_out: output(6144) ++ distance(1)
// ---------------------------------------------------------------------------
extern "C" void kernel_launch(void* const* d_in, const int* in_sizes, int n_in,
                              void* d_out, int out_size, void* d_ws, size_t ws_size,
                              hipStream_t stream) {
    (void)in_sizes; (void)n_in; (void)out_size; (void)ws_size;

    const float* x = (const float*)d_in[0];
    const float* Wp[8]; const float* Bp[8];
    for (int i = 0; i < 8; ++i) {
        Wp[i] = (const float*)d_in[1 + 2 * i];
        Bp[i] = (const float*)d_in[2 + 2 * i];
    }
    float* out  = (float*)d_out;
    float* ws   = (float*)d_ws;
    float* a0   = ws;
    float* a1   = ws + A_DIM;
    float* ylat = ws + 2 * A_DIM;

    dim3 blk(256);
    // encoder: 3x (6144<-6144, swish) then 4096<-6144 (linear) -> latent
    gemv_wmma_kernel<1><<<A_DIM / 32, blk, 0, stream>>>(Wp[0], Bp[0], x,    a0,   A_DIM);
    gemv_wmma_kernel<1><<<A_DIM / 32, blk, 0, stream>>>(Wp[1], Bp[1], a0,   a1,   A_DIM);
    gemv_wmma_kernel<1><<<A_DIM / 32, blk, 0, stream>>>(Wp[2], Bp[2], a1,   a0,   A_DIM);
    gemv_wmma_kernel<0><<<D_DIM / 32, blk, 0, stream>>>(Wp[3], Bp[3], a0,   ylat, A_DIM);
    // decoder: 6144<-4096 swish, 2x (6144<-6144 swish), 6144<-6144 + tanh
    gemv_wmma_kernel<1><<<A_DIM / 32, blk, 0, stream>>>(Wp[4], Bp[4], ylat, a0,   D_DIM);
    gemv_wmma_kernel<1><<<A_DIM / 32, blk, 0, stream>>>(Wp[5], Bp[5], a0,   a1,   A_DIM);
    gemv_wmma_kernel<1><<<A_DIM / 32, blk, 0, stream>>>(Wp[6], Bp[6], a1,   a0,   A_DIM);
    gemv_wmma_kernel<2><<<A_DIM / 32, blk, 0, stream>>>(Wp[7], Bp[7], a0,   out,  A_DIM);
    // KL distance tail
    distance_kernel<<<1, blk, 0, stream>>>(x, ylat, out + A_DIM);
}